// _Upsample2d_81655918231919
// MI455X (gfx1250) — compile-verified
//
#include <hip/hip_runtime.h>

typedef float v2f __attribute__((ext_vector_type(2)));
typedef float v8f __attribute__((ext_vector_type(8)));

#define NIMG 128          // B*C = 4*32
#define IH   256
#define IW   256
#define OH   512
#define OW   512
#define TPB  256          // 8 waves of 32
#define WAVES 8
#define TILES_TOTAL (NIMG * 32 * 32)   // 131072 16x16 output tiles

__device__ __forceinline__ int iabs(int v) { return v < 0 ? -v : v; }

__global__ __launch_bounds__(TPB)
void upsample2x_wmma(const float* __restrict__ x,
                     const float* __restrict__ f,
                     float* __restrict__ y) {
    // Shared: polyphase matrix W (16x16, stride 17), effective half-filter he[9],
    // and per-wave patch P / intermediate T (each 16x17 floats, wave-private).
    __shared__ float sHe[9];
    __shared__ float sW[16 * 17];
    __shared__ float sPT[WAVES][2 * 16 * 17];

    const int tid  = threadIdx.x;
    const int wave = tid >> 5;
    const int lane = tid & 31;
    const int lo   = lane & 15;     // M / N index within half-wave
    const int hi   = lane >> 4;     // 0: K = {0,1}, 1: K = {2,3} per WMMA chunk

    // ---- effective symmetric filter he[0..8] from the 15-tap f ----
    // (DCT-domain product == symmetric conv with he: he[0]=4f[7],
    //  he[j]=2(f[7+j]+f[6+j]), he[8]=2f[14], global scale 1/sqrt(5*pi))
    if (tid < 9) {
        float v;
        if (tid == 0)      v = 4.0f * f[7];
        else if (tid == 8) v = 2.0f * f[14];
        else               v = 2.0f * (f[7 + tid] + f[6 + tid]);
        sHe[tid] = v;
    }
    __syncthreads();

    // ---- banded polyphase matrix W[o][k]: one entry per thread ----
    {
        const int o = tid >> 4, k = tid & 15;
        const int a = o & 1, rl = o >> 1;
        const int s = k - 4 - rl;
        float v = 0.0f;
        if (a == 0) {                      // even phase: 9 taps he[0,2,..,8]
            if (s >= -4 && s <= 4) v = sHe[2 * iabs(s)];
        } else {                           // odd phase: 8 taps he[1,3,..,7]
            if (s >= -3 && s <= 4) v = sHe[iabs(2 * s - 1)];
        }
        sW[o * 17 + k] = v;
    }
    __syncthreads();   // cross-wave: all waves read sW below

    // This lane's W-row entries (used as both stage-1 B = W^T column and
    // stage-2 A = W row): wa[2c] = W[lo][4c + 2*hi], wa[2c+1] = next K.
    float wa[8];
#pragma unroll
    for (int c = 0; c < 4; ++c) {
        wa[2 * c]     = sW[lo * 17 + 4 * c + 2 * hi];
        wa[2 * c + 1] = sW[lo * 17 + 4 * c + 2 * hi + 1];
    }

    // ---- tile coordinates ----
    const int gtile = blockIdx.x * WAVES + wave;
    const int img = gtile >> 10;
    const int rem = gtile & 1023;
    const int tI  = rem >> 5;
    const int tJ  = rem & 31;

    float* P = &sPT[wave][0];
    float* T = &sPT[wave][16 * 17];

    // ---- load 16x16 input patch (rows/cols 8*t-1 .. 8*t+14 in padded coords) ----
    const float* xi = x + (size_t)img * (IH * IW);
    const int gi0 = 8 * tI - 1;   // padded-row index of patch row 0 (range [-1,262])
    const int gj0 = 8 * tJ - 1;
#pragma unroll
    for (int i = 0; i < 8; ++i) {
        const int row = hi + 2 * i;       // element e = lane + 32*i -> (e>>4, e&15)
        const int pi = gi0 + row;
        const int pj = gj0 + lo;
        float v = 0.0f;
        if (pi >= 0 && pi <= 261 && pj >= 0 && pj <= 261) {  // pi/pj == -1 or 262 -> 0
            int r = pi - 3; r = r < 0 ? -r : (r > 255 ? 510 - r : r);   // reflect pad 3
            int c = pj - 3; c = c < 0 ? -c : (c > 255 ? 510 - c : c);
            v = xi[r * IW + c];
        }
        P[row * 17 + lo] = v;
    }
    // No block barrier: P is wave-private; same-wave LDS ops are in-order and
    // the compiler inserts s_wait_dscnt for the RAW before the reads below.

    // ---- stage 1: T = P x W^T  (four chained 16x16x4 f32 WMMAs) ----
    v8f acc = {0.f, 0.f, 0.f, 0.f, 0.f, 0.f, 0.f, 0.f};
#pragma unroll
    for (int kk = 0; kk < 16; kk += 4) {
        const int kb = kk + 2 * hi;
        v2f a, b;
        a.x = P[lo * 17 + kb];            // A[m=lo][kb],  A = P
        a.y = P[lo * 17 + kb + 1];
        b.x = wa[(kk >> 1) + 0];          // B[k][n=lo] = W[lo][k]
        b.y = wa[(kk >> 1) + 1];
        acc = __builtin_amdgcn_wmma_f32_16x16x4_f32(
            false, a, false, b, (short)0, acc, false, false);
    }

    // D-layout -> LDS so stage 2 can read it in B-operand layout (wave-private)
#pragma unroll
    for (int v = 0; v < 8; ++v)
        T[(v + 8 * hi) * 17 + lo] = acc[v];

    // ---- stage 2: Out = W x T ----
    v8f out = {0.f, 0.f, 0.f, 0.f, 0.f, 0.f, 0.f, 0.f};
#pragma unroll
    for (int kk = 0; kk < 16; kk += 4) {
        const int kb = kk + 2 * hi;
        v2f a, b;
        a.x = wa[(kk >> 1) + 0];          // A[m=lo][k] = W[lo][k]
        a.y = wa[(kk >> 1) + 1];
        b.x = T[kb * 17 + lo];            // B[k][n=lo] = T[k][lo]
        b.y = T[(kb + 1) * 17 + lo];
        out = __builtin_amdgcn_wmma_f32_16x16x4_f32(
            false, a, false, b, (short)0, out, false, false);
    }

    // ---- store 16x16 output tile, scaled by 1/sqrt(5*pi) ----
    // Output is write-once / never re-read by this kernel: use non-temporal
    // stores (TH_NT) so the 134 MB result doesn't evict the 33.5 MB input
    // working set (which fits in the 192 MB L2 and feeds the halo re-reads).
    const float invs = 0.25231325f;       // 1/sqrt(5*pi)
    float* yo = y + (size_t)img * (OH * OW) + (size_t)(16 * tI) * OW + 16 * tJ;
#pragma unroll
    for (int v = 0; v < 8; ++v)
        __builtin_nontemporal_store(out[v] * invs, &yo[(v + 8 * hi) * OW + lo]);
}

extern "C" void kernel_launch(void* const* d_in, const int* in_sizes, int n_in,
                              void* d_out, int out_size, void* d_ws, size_t ws_size,
                              hipStream_t stream) {
    const float* x = (const float*)d_in[0];   // (4,32,256,256) f32
    const float* f = (const float*)d_in[1];   // 15-tap filter f32
    float* y = (float*)d_out;                 // (4,32,512,512) f32
    (void)in_sizes; (void)n_in; (void)out_size; (void)d_ws; (void)ws_size;
    upsample2x_wmma<<<TILES_TOTAL / WAVES, TPB, 0, stream>>>(x, f, y);
}